// GAT_21260088115447
// MI455X (gfx1250) — compile-verified
//
#include <hip/hip_runtime.h>
#include <hip/hip_bf16.h>
#include <math.h>

typedef float v2f __attribute__((ext_vector_type(2)));
typedef float v8f __attribute__((ext_vector_type(8)));

#define NEG_SLOPE 0.2f

#if __has_builtin(__builtin_amdgcn_global_load_async_to_lds_b128) && \
    __has_builtin(__builtin_amdgcn_s_wait_asynccnt)
#define STAGE_ASYNC 1
// async copy builtin takes (AS1 int4*, AS3 int4*, imm offset, imm cpol)
typedef int i32x4 __attribute__((vector_size(4 * sizeof(int))));
typedef __attribute__((address_space(1))) i32x4 g_i32x4;   // global
typedef __attribute__((address_space(3))) i32x4 l_i32x4;   // LDS
#else
#define STAGE_ASYNC 0
#endif

// ---------------------------------------------------------------- utilities

__device__ __forceinline__ float lrelu(float v) {
    return v >= 0.f ? v : NEG_SLOPE * v;
}

// ordered-float atomic max: signed-int max handles {+,mixed}, unsigned min handles {-}
__device__ __forceinline__ void atomicMaxF(float* addr, float v) {
    if (v >= 0.f)
        atomicMax((int*)addr, __float_as_int(v));
    else
        atomicMin((unsigned int*)addr, __float_as_uint(v));
}

// resolve edge (src,dst); edges [E, E+N) are the appended self-loops
__device__ __forceinline__ void edge_sd(const int* __restrict__ src,
                                        const int* __restrict__ dst,
                                        int ei, int E, int& s, int& d) {
    if (ei < E) { s = src[ei]; d = dst[ei]; }
    else        { s = d = ei - E; }
}

__global__ void fill_f32(float* __restrict__ p, float v, int n) {
    int i = blockIdx.x * blockDim.x + threadIdx.x;
    if (i < n) p[i] = v;
}

// ---------------------------------------------------------------- GEMM1: h1 = x @ W1   [N,256]x[256,64]
// W1 (64 KB) is staged once per workgroup into LDS via the CDNA5 async
// global->LDS copy path (ASYNCcnt), then each wave computes a 16-row node
// block as 4 column tiles of 16 with f32 WMMA, K stepped by 4.

__global__ void gemm1_wmma(const float* __restrict__ x, const float* __restrict__ W,
                           float* __restrict__ h, int N) {
    __shared__ __align__(16) float sW[256 * 64];   // 64 KB of the 320 KB WGP LDS
    {
        const int tid = threadIdx.x;               // 256 threads stage 4096 float4s
#if STAGE_ASYNC
        #pragma unroll
        for (int i = 0; i < 16; ++i) {
            int f4 = tid + i * 256;
            __builtin_amdgcn_global_load_async_to_lds_b128(
                (g_i32x4*)(W + (size_t)f4 * 4),
                (l_i32x4*)(sW + (size_t)f4 * 4),
                0, 0);
        }
        __builtin_amdgcn_s_wait_asynccnt(0);       // own async writes visible in LDS
#else
        #pragma unroll
        for (int i = 0; i < 16; ++i) {
            int f4 = tid + i * 256;
            ((float4*)sW)[f4] = ((const float4*)W)[f4];
        }
#endif
        __syncthreads();
    }

    int wave = (blockIdx.x * blockDim.x + threadIdx.x) >> 5;
    int lane = threadIdx.x & 31;
    int row0 = wave << 4;
    if (row0 >= N) return;

    int half = lane >> 4;     // 0: lanes 0-15, 1: lanes 16-31
    int l    = lane & 15;
    int rr   = row0 + l; if (rr >= N) rr = N - 1;
    const float* xrow = x + (size_t)rr * 256;

    // warm the row (8 cachelines of 128B) -> global_prefetch_b8
    #pragma unroll
    for (int pf = 0; pf < 8; ++pf)
        __builtin_prefetch(xrow + pf * 32, 0, 1);

    v8f acc0 = {}, acc1 = {}, acc2 = {}, acc3 = {};
    for (int k0 = 0; k0 < 256; k0 += 4) {
        // A 16x4: v0 = K=(half*2), v1 = K=(half*2+1)
        v2f a;
        a.x = xrow[k0 + half * 2 + 0];
        a.y = xrow[k0 + half * 2 + 1];
        const float* w0 = sW + (k0 + half * 2 + 0) * 64;
        const float* w1 = sW + (k0 + half * 2 + 1) * 64;
        v2f b0, b1, b2, b3;
        b0.x = w0[l +  0]; b0.y = w1[l +  0];
        b1.x = w0[l + 16]; b1.y = w1[l + 16];
        b2.x = w0[l + 32]; b2.y = w1[l + 32];
        b3.x = w0[l + 48]; b3.y = w1[l + 48];
        acc0 = __builtin_amdgcn_wmma_f32_16x16x4_f32(false, a, false, b0, (short)0, acc0, false, false);
        acc1 = __builtin_amdgcn_wmma_f32_16x16x4_f32(false, a, false, b1, (short)0, acc1, false, false);
        acc2 = __builtin_amdgcn_wmma_f32_16x16x4_f32(false, a, false, b2, (short)0, acc2, false, false);
        acc3 = __builtin_amdgcn_wmma_f32_16x16x4_f32(false, a, false, b3, (short)0, acc3, false, false);
    }
    // C: VGPR r holds row (row0 + r + half*8), col l (per 16-wide tile)
    #pragma unroll
    for (int r = 0; r < 8; ++r) {
        int row = row0 + r + half * 8;
        if (row < N) {
            float* o = h + (size_t)row * 64;
            o[l +  0] = acc0[r];
            o[l + 16] = acc1[r];
            o[l + 32] = acc2[r];
            o[l + 48] = acc3[r];
        }
    }
}

// ---------------------------------------------------------------- GEMM2: h2 = hrelu @ W2   [N,64]x[64,40]

__global__ void gemm2_wmma(const float* __restrict__ A, const float* __restrict__ W,
                           float* __restrict__ out, int N) {
    __shared__ __align__(16) float sW[64 * 40];    // 10 KB
    {
        const int tid = threadIdx.x;               // 640 float4s total
#if STAGE_ASYNC
        #pragma unroll
        for (int i = 0; i < 3; ++i) {
            int f4 = tid + i * 256;
            if (f4 < 640)
                __builtin_amdgcn_global_load_async_to_lds_b128(
                    (g_i32x4*)(W + (size_t)f4 * 4),
                    (l_i32x4*)(sW + (size_t)f4 * 4),
                    0, 0);
        }
        __builtin_amdgcn_s_wait_asynccnt(0);
#else
        #pragma unroll
        for (int i = 0; i < 3; ++i) {
            int f4 = tid + i * 256;
            if (f4 < 640)
                ((float4*)sW)[f4] = ((const float4*)W)[f4];
        }
#endif
        __syncthreads();
    }

    int wave = (blockIdx.x * blockDim.x + threadIdx.x) >> 5;
    int lane = threadIdx.x & 31;
    int row0 = wave << 4;
    if (row0 >= N) return;

    int half = lane >> 4;
    int l    = lane & 15;
    int rr   = row0 + l; if (rr >= N) rr = N - 1;
    const float* arow = A + (size_t)rr * 64;

    #pragma unroll
    for (int pf = 0; pf < 2; ++pf)
        __builtin_prefetch(arow + pf * 32, 0, 1);

    v8f acc0 = {}, acc1 = {}, acc2 = {};
    for (int k0 = 0; k0 < 64; k0 += 4) {
        v2f a;
        a.x = arow[k0 + half * 2 + 0];
        a.y = arow[k0 + half * 2 + 1];
        const float* w0 = sW + (k0 + half * 2 + 0) * 40;
        const float* w1 = sW + (k0 + half * 2 + 1) * 40;
        v2f b0, b1, b2;
        int c1 = l + 16, c2 = l + 32;
        b0.x = w0[l];  b0.y = w1[l];
        b1.x = w0[c1]; b1.y = w1[c1];
        b2.x = (c2 < 40) ? w0[c2] : 0.f;
        b2.y = (c2 < 40) ? w1[c2] : 0.f;
        acc0 = __builtin_amdgcn_wmma_f32_16x16x4_f32(false, a, false, b0, (short)0, acc0, false, false);
        acc1 = __builtin_amdgcn_wmma_f32_16x16x4_f32(false, a, false, b1, (short)0, acc1, false, false);
        acc2 = __builtin_amdgcn_wmma_f32_16x16x4_f32(false, a, false, b2, (short)0, acc2, false, false);
    }
    #pragma unroll
    for (int r = 0; r < 8; ++r) {
        int row = row0 + r + half * 8;
        if (row < N) {
            float* o = out + (size_t)row * 40;
            o[l]      = acc0[r];
            o[l + 16] = acc1[r];
            if (l + 32 < 40) o[l + 32] = acc2[r];
        }
    }
}

// ---------------------------------------------------------------- per-node attention logits

__global__ void alpha_k(const float* __restrict__ h, const float* __restrict__ asrc,
                        const float* __restrict__ adst, float* __restrict__ aso,
                        float* __restrict__ ado, int N, int H, int C) {
    int t = blockIdx.x * blockDim.x + threadIdx.x;
    if (t >= N * H) return;
    int n = t / H, hh = t - n * H;
    const float* hp = h + (size_t)n * H * C + hh * C;
    const float* ws = asrc + hh * C;
    const float* wd = adst + hh * C;
    float s1 = 0.f, s2 = 0.f;
    for (int c = 0; c < C; ++c) {
        float v = hp[c];
        s1 = fmaf(v, ws[c], s1);
        s2 = fmaf(v, wd[c], s2);
    }
    aso[t] = s1;
    ado[t] = s2;
}

// ---------------------------------------------------------------- edge passes (softmax over incoming edges per dst)

__global__ void edge_max_k(const int* __restrict__ src, const int* __restrict__ dst,
                           const float* __restrict__ as_, const float* __restrict__ ad_,
                           float* __restrict__ m_, int E, int N, int H) {
    int t = blockIdx.x * blockDim.x + threadIdx.x;
    int total = (E + N) * H;
    if (t >= total) return;
    int h = t % H, ei = t / H;
    int s, d; edge_sd(src, dst, ei, E, s, d);
    float v = lrelu(as_[s * H + h] + ad_[d * H + h]);
    atomicMaxF(&m_[d * H + h], v);
}

__global__ void edge_expsum_k(const int* __restrict__ src, const int* __restrict__ dst,
                              const float* __restrict__ as_, const float* __restrict__ ad_,
                              const float* __restrict__ m_, float* __restrict__ dn_,
                              int E, int N, int H) {
    int t = blockIdx.x * blockDim.x + threadIdx.x;
    int total = (E + N) * H;
    if (t >= total) return;
    int h = t % H, ei = t / H;
    int s, d; edge_sd(src, dst, ei, E, s, d);
    float v = lrelu(as_[s * H + h] + ad_[d * H + h]);
    unsafeAtomicAdd(&dn_[d * H + h], __expf(v - m_[d * H + h]));
}

// layer-1 aggregation: H=8, C=8; one thread per (edge, head) handles its 8 channels
__global__ void edge_agg1_k(const int* __restrict__ src, const int* __restrict__ dst,
                            const float* __restrict__ as_, const float* __restrict__ ad_,
                            const float* __restrict__ m_, const float* __restrict__ dn_,
                            const float* __restrict__ h1, float* __restrict__ agg,
                            int E, int N) {
    int t = blockIdx.x * blockDim.x + threadIdx.x;
    int total = (E + N) * 8;
    if (t >= total) return;
    int h = t & 7, ei = t >> 3;
    int s, d; edge_sd(src, dst, ei, E, s, d);
    float v = lrelu(as_[s * 8 + h] + ad_[d * 8 + h]);
    float a = __expf(v - m_[d * 8 + h]) / dn_[d * 8 + h];
    const float* hs = h1 + (size_t)s * 64 + h * 8;
    float* ag = agg + (size_t)d * 64 + h * 8;
    #pragma unroll
    for (int c = 0; c < 8; ++c)
        unsafeAtomicAdd(&ag[c], hs[c] * a);
}

// layer-2 aggregation: H=1, C=40; one thread per (edge, 5-channel chunk)
__global__ void edge_agg2_k(const int* __restrict__ src, const int* __restrict__ dst,
                            const float* __restrict__ as_, const float* __restrict__ ad_,
                            const float* __restrict__ m_, const float* __restrict__ dn_,
                            const float* __restrict__ h2, float* __restrict__ out,
                            int E, int N) {
    int t = blockIdx.x * blockDim.x + threadIdx.x;
    int total = (E + N) * 8;
    if (t >= total) return;
    int c0 = (t & 7) * 5, ei = t >> 3;
    int s, d; edge_sd(src, dst, ei, E, s, d);
    float v = lrelu(as_[s] + ad_[d]);
    float a = __expf(v - m_[d]) / dn_[d];
    const float* hs = h2 + (size_t)s * 40 + c0;
    float* ag = out + (size_t)d * 40 + c0;
    #pragma unroll
    for (int c = 0; c < 5; ++c)
        unsafeAtomicAdd(&ag[c], hs[c] * a);
}

// ---------------------------------------------------------------- pointwise epilogues

__global__ void bias_relu_k(float* __restrict__ p, const float* __restrict__ b, int n, int ch) {
    int i = blockIdx.x * blockDim.x + threadIdx.x;
    if (i >= n) return;
    int c = i % ch;
    p[i] = fmaxf(p[i] + b[c], 0.f);
}

__global__ void logsoftmax_bias_k(float* __restrict__ out, const float* __restrict__ b, int N) {
    int n = blockIdx.x * blockDim.x + threadIdx.x;
    if (n >= N) return;
    float* row = out + (size_t)n * 40;
    float v[40];
    float mx = -INFINITY;
    #pragma unroll
    for (int c = 0; c < 40; ++c) {
        v[c] = row[c] + b[c];
        mx = fmaxf(mx, v[c]);
    }
    float s = 0.f;
    #pragma unroll
    for (int c = 0; c < 40; ++c) s += __expf(v[c] - mx);
    float ls = __logf(s) + mx;
    #pragma unroll
    for (int c = 0; c < 40; ++c) row[c] = v[c] - ls;
}

// ---------------------------------------------------------------- launch

extern "C" void kernel_launch(void* const* d_in, const int* in_sizes, int n_in,
                              void* d_out, int out_size, void* d_ws, size_t ws_size,
                              hipStream_t stream) {
    const float* x   = (const float*)d_in[0];
    const int*   ei  = (const int*)d_in[1];
    const float* W1  = (const float*)d_in[2];
    const float* a1s = (const float*)d_in[3];
    const float* a1d = (const float*)d_in[4];
    const float* b1  = (const float*)d_in[5];
    const float* W2  = (const float*)d_in[6];
    const float* a2s = (const float*)d_in[7];
    const float* a2d = (const float*)d_in[8];
    const float* b2  = (const float*)d_in[9];
    float* out = (float*)d_out;

    const int N = in_sizes[0] / 256;
    const int E = in_sizes[1] / 2;
    const int ET = E + N;                 // with self-loops
    const int* src = ei;
    const int* dst = ei + E;

    // workspace layout (floats); layer-2 reuses layer-1 regions
    float* ws  = (float*)d_ws;
    float* h1  = ws;                        // N*64  (layer2: h2 uses N*40 of it)
    float* as_ = h1  + (size_t)N * 64;      // N*8   (layer2: N)
    float* ad_ = as_ + (size_t)N * 8;       // N*8
    float* m_  = ad_ + (size_t)N * 8;       // N*8
    float* dn_ = m_  + (size_t)N * 8;       // N*8
    float* agg = dn_ + (size_t)N * 8;       // N*64  (relu'd in place -> layer2 input)

    const int B = 256;
    auto blocks = [](int n, int b) { return (n + b - 1) / b; };
    int gemm_blocks = blocks(blocks(N, 16), 8);   // 8 waves/block, 16 rows/wave

    // ---------- layer 1 (H=8, C=8) ----------
    fill_f32<<<blocks(N * 8, B), B, 0, stream>>>(m_, -INFINITY, N * 8);
    fill_f32<<<blocks(N * 8, B), B, 0, stream>>>(dn_, 0.f, N * 8);
    fill_f32<<<blocks(N * 64, B), B, 0, stream>>>(agg, 0.f, N * 64);

    gemm1_wmma<<<gemm_blocks, B, 0, stream>>>(x, W1, h1, N);
    alpha_k<<<blocks(N * 8, B), B, 0, stream>>>(h1, a1s, a1d, as_, ad_, N, 8, 8);

    edge_max_k   <<<blocks(ET * 8, B), B, 0, stream>>>(src, dst, as_, ad_, m_, E, N, 8);
    edge_expsum_k<<<blocks(ET * 8, B), B, 0, stream>>>(src, dst, as_, ad_, m_, dn_, E, N, 8);
    edge_agg1_k  <<<blocks(ET * 8, B), B, 0, stream>>>(src, dst, as_, ad_, m_, dn_, h1, agg, E, N);

    bias_relu_k<<<blocks(N * 64, B), B, 0, stream>>>(agg, b1, N * 64, 64);

    // ---------- layer 2 (H=1, C=40) ----------
    float* h2  = h1;    // reuse
    fill_f32<<<blocks(N, B), B, 0, stream>>>(m_, -INFINITY, N);
    fill_f32<<<blocks(N, B), B, 0, stream>>>(dn_, 0.f, N);
    fill_f32<<<blocks(N * 40, B), B, 0, stream>>>(out, 0.f, N * 40);

    gemm2_wmma<<<gemm_blocks, B, 0, stream>>>(agg, W2, h2, N);
    alpha_k<<<blocks(N, B), B, 0, stream>>>(h2, a2s, a2d, as_, ad_, N, 1, 40);

    edge_max_k   <<<blocks(ET, B), B, 0, stream>>>(src, dst, as_, ad_, m_, E, N, 1);
    edge_expsum_k<<<blocks(ET, B), B, 0, stream>>>(src, dst, as_, ad_, m_, dn_, E, N, 1);
    edge_agg2_k  <<<blocks(ET * 8, B), B, 0, stream>>>(src, dst, as_, ad_, m_, dn_, h2, out, E, N);

    logsoftmax_bias_k<<<blocks(N, B), B, 0, stream>>>(out, b2, N);
}